// SGC_Net_4320737100481
// MI455X (gfx1250) — compile-verified
//
#include <hip/hip_runtime.h>
#include <math.h>

typedef __attribute__((ext_vector_type(2))) float v2f;
typedef __attribute__((ext_vector_type(8))) float v8f;

#define N_NODES   150000
#define N_EDGES   1200000
#define N_FEAT    64
#define N_CLASSES 40
#define ROW_TILES (N_NODES / 16)   // 9375, exact

// ---------------- degree / normalization ----------------

__global__ void deg_init_kernel(float* __restrict__ deg) {
    int i = blockIdx.x * blockDim.x + threadIdx.x;
    if (i < N_NODES) deg[i] = 1.0f;  // self-loop contributes 1 to every node's degree
}

__global__ void deg_accum_kernel(const long long* __restrict__ col, float* __restrict__ deg) {
    int e = blockIdx.x * blockDim.x + threadIdx.x;
    if (e < N_EDGES) atomicAdd(&deg[(int)col[e]], 1.0f);
}

__global__ void dinv_kernel(float* __restrict__ deg) {
    int i = blockIdx.x * blockDim.x + threadIdx.x;
    if (i < N_NODES) {
        float d = deg[i];
        deg[i] = (d > 0.0f) ? rsqrtf(d) : 0.0f;
    }
}

// ---------------- SpMM hop ----------------
// h_out[i] = dinv[i]^2 * h_in[i]   (self-loop term; also serves as the zero-init)
__global__ void hop_init_kernel(const float* __restrict__ hin,
                                const float* __restrict__ dinv,
                                float* __restrict__ hout) {
    int t = blockIdx.x * blockDim.x + threadIdx.x;   // N_NODES * 16 float4 groups
    if (t < N_NODES * (N_FEAT / 4)) {
        int node = t >> 4;
        float s = dinv[node];
        s = s * s;
        float4 v = reinterpret_cast<const float4*>(hin)[t];
        float4 o;
        o.x = s * v.x; o.y = s * v.y; o.z = s * v.z; o.w = s * v.w;
        reinterpret_cast<float4*>(hout)[t] = o;
    }
}

// h_out[c] += dinv[r]*dinv[c] * h_in[r]  for every edge; 16 threads per edge (float4 each)
__global__ void hop_scatter_kernel(const long long* __restrict__ row,
                                   const long long* __restrict__ col,
                                   const float* __restrict__ dinv,
                                   const float* __restrict__ hin,
                                   float* __restrict__ hout) {
    int tid = blockIdx.x * blockDim.x + threadIdx.x;   // N_EDGES * 16
    if (tid < N_EDGES * (N_FEAT / 4)) {
        int e = tid >> 4;
        int f = (tid & 15) << 2;
        int r = (int)row[e];
        int c = (int)col[e];
        float w = dinv[r] * dinv[c];
        float4 v = *reinterpret_cast<const float4*>(hin + (size_t)r * N_FEAT + f);
        float* dst = hout + (size_t)c * N_FEAT + f;
        atomicAdd(dst + 0, w * v.x);
        atomicAdd(dst + 1, w * v.y);
        atomicAdd(dst + 2, w * v.z);
        atomicAdd(dst + 3, w * v.w);
    }
}

// ---------------- dense classifier: h @ W^T + b, log_softmax ----------------
// One wave handles a 16-row tile of h. K=64 chained as 16 WMMA f32 16x16x4 steps,
// N=40 padded to 3 tiles of 16. D tiles staged in LDS, then lanes 0..15 each
// finish log-softmax for one row.
__global__ __launch_bounds__(256) void classify_kernel(const float* __restrict__ h,
                                                       const float* __restrict__ W,
                                                       const float* __restrict__ bias,
                                                       float* __restrict__ out) {
    __shared__ float lds[8][16][49];   // [wave][row][col], 49-stride avoids bank conflicts

    int lane = threadIdx.x & 31;
    int wave = threadIdx.x >> 5;
    int rowBlock = blockIdx.x * 8 + wave;
    bool active = rowBlock < ROW_TILES;   // wave-uniform: EXEC all-1s inside

    int half = lane >> 4;   // which 16-lane half
    int m    = lane & 15;   // row index (A) / col index (B) within tile

    if (active) {
        v8f acc0 = {}, acc1 = {}, acc2 = {};
        const float* hrow = h + (size_t)(rowBlock * 16 + m) * N_FEAT;

        // B row pointers for the 3 column tiles. Tile 2 columns >= N_CLASSES are
        // padding: load from a clamped (always-valid) row and multiply by 0 —
        // keeps the loads unconditional (no exec-mask branches in the inner loop).
        int n2 = 32 + m;
        const float* w0row = W + (size_t)m * N_FEAT;
        const float* w1row = W + (size_t)(16 + m) * N_FEAT;
        const float* w2row = W + (size_t)((n2 < N_CLASSES) ? n2 : 0) * N_FEAT;
        float pad2 = (n2 < N_CLASSES) ? 1.0f : 0.0f;

        for (int kk = 0; kk < 16; ++kk) {
            int k0 = kk * 4 + half * 2;
            // A fragment (16x4 f32): VGPR0={K0,K2}, VGPR1={K1,K3} across lane halves
            v2f a;
            a.x = hrow[k0];
            a.y = hrow[k0 + 1];
            // B fragments mirror A's K striping; N = lane&15
            v2f b0, b1, b2;
            b0.x = w0row[k0];        b0.y = w0row[k0 + 1];
            b1.x = w1row[k0];        b1.y = w1row[k0 + 1];
            b2.x = pad2 * w2row[k0]; b2.y = pad2 * w2row[k0 + 1];

            acc0 = __builtin_amdgcn_wmma_f32_16x16x4_f32(false, a, false, b0, (short)0, acc0, false, false);
            acc1 = __builtin_amdgcn_wmma_f32_16x16x4_f32(false, a, false, b1, (short)0, acc1, false, false);
            acc2 = __builtin_amdgcn_wmma_f32_16x16x4_f32(false, a, false, b2, (short)0, acc2, false, false);
        }

        // D layout: acc[v], lane l -> row = v + 8*(l>>4), col = (l&15)
        #pragma unroll
        for (int v = 0; v < 8; ++v) {
            int r = v + 8 * half;
            lds[wave][r][ 0 + m] = acc0[v];
            lds[wave][r][16 + m] = acc1[v];
            lds[wave][r][32 + m] = acc2[v];
        }
    }

    __syncthreads();

    if (active && lane < 16) {
        int row = lane;
        // pass 1: max
        float mx = -3.0e38f;
        #pragma unroll
        for (int c = 0; c < N_CLASSES; ++c) {
            float t = lds[wave][row][c] + bias[c];
            mx = fmaxf(mx, t);
        }
        // pass 2: sum exp
        float s = 0.0f;
        #pragma unroll
        for (int c = 0; c < N_CLASSES; ++c) {
            float t = lds[wave][row][c] + bias[c];
            s += __expf(t - mx);
        }
        float lse = mx + __logf(s);
        // pass 3: write
        float* o = out + (size_t)(rowBlock * 16 + row) * N_CLASSES;
        #pragma unroll
        for (int c = 0; c < N_CLASSES; ++c) {
            float t = lds[wave][row][c] + bias[c];
            o[c] = t - lse;
        }
    }
}

// ---------------- launch ----------------

extern "C" void kernel_launch(void* const* d_in, const int* in_sizes, int n_in,
                              void* d_out, int out_size, void* d_ws, size_t ws_size,
                              hipStream_t stream) {
    const float*     x  = (const float*)d_in[0];
    const long long* ei = (const long long*)d_in[1];   // int64 edge_index [2, E]
    const float*     W  = (const float*)d_in[2];       // [40, 64]
    const float*     b  = (const float*)d_in[3];       // [40]
    float* out = (float*)d_out;

    const long long* row = ei;             // sources
    const long long* col = ei + N_EDGES;   // destinations

    char* ws = (char*)d_ws;
    float* dinv = (float*)ws;                                        // N_NODES floats
    float* h1   = (float*)(ws + (size_t)N_NODES * sizeof(float));    // offset 600000 B (16-aligned)
    float* h2   = h1 + (size_t)N_NODES * N_FEAT;

    const int T = 256;
    // degree + D^{-1/2}
    deg_init_kernel <<<(N_NODES + T - 1) / T, T, 0, stream>>>(dinv);
    deg_accum_kernel<<<(N_EDGES + T - 1) / T, T, 0, stream>>>(col, dinv);
    dinv_kernel     <<<(N_NODES + T - 1) / T, T, 0, stream>>>(dinv);

    const int nodeVec = N_NODES * (N_FEAT / 4);
    const int edgeVec = N_EDGES * (N_FEAT / 4);

    // hop 1: h1 = norm-SpMM(x)
    hop_init_kernel   <<<(nodeVec + T - 1) / T, T, 0, stream>>>(x, dinv, h1);
    hop_scatter_kernel<<<(edgeVec + T - 1) / T, T, 0, stream>>>(row, col, dinv, x, h1);
    // hop 2: h2 = norm-SpMM(h1)
    hop_init_kernel   <<<(nodeVec + T - 1) / T, T, 0, stream>>>(h1, dinv, h2);
    hop_scatter_kernel<<<(edgeVec + T - 1) / T, T, 0, stream>>>(row, col, dinv, h1, h2);

    // classifier + log_softmax
    const int blocks = (ROW_TILES + 7) / 8;   // 8 waves (16-row tiles) per block
    classify_kernel<<<blocks, T, 0, stream>>>(h2, W, b, out);
}